// SAGENet_48541720379895
// MI455X (gfx1250) — compile-verified
//
#include <hip/hip_runtime.h>

typedef __attribute__((ext_vector_type(2))) float v2f;
typedef __attribute__((ext_vector_type(8))) float v8f;

#define N_NODES 100000
#define EMB_DIM 128
#define HIDDEN  128
#define OUT_C   64
#define N_EDGES 800000

// ---------------------------------------------------------------------------
// Zero a float buffer (float4 granularity, grid-stride).
// ---------------------------------------------------------------------------
__global__ void zero4_kernel(float4* __restrict__ p, int n4) {
    int i = blockIdx.x * blockDim.x + threadIdx.x;
    int stride = gridDim.x * blockDim.x;
    float4 z = make_float4(0.f, 0.f, 0.f, 0.f);
    for (; i < n4; i += stride) p[i] = z;
}

// ---------------------------------------------------------------------------
// Degree count: cnt[dst[e]] += 1  (f32 atomics, L2-resident)
// ei layout: [2, N_EDGES] -> src = ei[e], dst = ei[N_EDGES + e]
// ---------------------------------------------------------------------------
__global__ void degree_kernel(const int* __restrict__ ei, float* __restrict__ cnt) {
    int e = blockIdx.x * blockDim.x + threadIdx.x;
    if (e < N_EDGES) atomicAdd(&cnt[ei[N_EDGES + e]], 1.0f);
}

// ---------------------------------------------------------------------------
// Edge scatter: agg[dst] += x[(xidx ? xidx[src] : src)]
// One thread handles a 4-float chunk of one edge message.
// F = feature dim (128 for both layers here).
// ---------------------------------------------------------------------------
template<int F>
__global__ void scatter_kernel(const int* __restrict__ ei,
                               const float* __restrict__ x,
                               const int* __restrict__ xidx,
                               float* __restrict__ agg) {
    const int CH = F / 4;                      // chunks per edge
    int t = blockIdx.x * blockDim.x + threadIdx.x;
    if (t >= N_EDGES * CH) return;
    int e = t / CH;
    int f = (t - e * CH) * 4;
    int s = ei[e];
    int d = ei[N_EDGES + e];
    if (xidx) s = xidx[s];
    const float4 v = *reinterpret_cast<const float4*>(x + (size_t)s * F + f);
    float* a = agg + (size_t)d * F + f;
    atomicAdd(a + 0, v.x);
    atomicAdd(a + 1, v.y);
    atomicAdd(a + 2, v.z);
    atomicAdd(a + 3, v.w);
}

// ---------------------------------------------------------------------------
// Fused SAGE layer:  out = act( (agg/max(cnt,1)) @ Wl + bias + X @ Wr )
//
// One wave32 computes one 16x16 output tile using V_WMMA_F32_16X16X4_F32.
// Operand layouts per CDNA5 ISA 7.12.2:
//   A (16x4 f32):  lanes 0-15 hold M=lane (K=0,1), lanes 16-31 hold M=lane-16
//                  (K=2,3); VGPR0/1 = first/second K of the pair.
//   B (4x16 f32):  lanes 0-15 hold N=lane, lane-half selects K pair.
//   C/D (16x16):   VGPR v: lanes 0-15 -> M=v, lanes 16-31 -> M=v+8; N=lane&15.
// Block = (32,4): 4 waves cover 4 consecutive column tiles of one row tile.
// ---------------------------------------------------------------------------
template<int K, int OUT, bool RELU>
__global__ __launch_bounds__(128)
void sage_wmma_kernel(const float* __restrict__ agg,
                      const float* __restrict__ cnt,
                      const float* __restrict__ X,
                      const int* __restrict__ xidx,
                      const float* __restrict__ Wl,
                      const float* __restrict__ bias,
                      const float* __restrict__ Wr,
                      float* __restrict__ out) {
    const int lane = threadIdx.x;            // 0..31
    const int wave = threadIdx.y;            // 0..3
    const int rowt = blockIdx.x;             // 16-row tile
    const int colt = blockIdx.y * 4 + wave;  // 16-col tile
    const int half = lane >> 4;              // 0 or 1
    const int l16  = lane & 15;

    const int arow = rowt * 16 + l16;        // A-operand row for this lane
    const int col  = colt * 16 + l16;        // B/C column for this lane

    const float inv = 1.0f / fmaxf(cnt[arow], 1.0f);
    int xrow = arow;
    if (xidx) xrow = xidx[arow];

    const float* __restrict__ aA = agg + (size_t)arow * K;
    const float* __restrict__ aX = X   + (size_t)xrow * K;

    v8f c = {0.f, 0.f, 0.f, 0.f, 0.f, 0.f, 0.f, 0.f};

    // mean @ Wl
    #pragma unroll
    for (int k = 0; k < K; k += 4) {
        const int ko = k + half * 2;
        v2f a; a.x = aA[ko] * inv;              a.y = aA[ko + 1] * inv;
        v2f b; b.x = Wl[(size_t)ko * OUT + col]; b.y = Wl[(size_t)(ko + 1) * OUT + col];
        c = __builtin_amdgcn_wmma_f32_16x16x4_f32(false, a, false, b,
                                                  (short)0, c, false, false);
    }
    // X @ Wr (accumulate into same C)
    #pragma unroll
    for (int k = 0; k < K; k += 4) {
        const int ko = k + half * 2;
        v2f a; a.x = aX[ko];                    a.y = aX[ko + 1];
        v2f b; b.x = Wr[(size_t)ko * OUT + col]; b.y = Wr[(size_t)(ko + 1) * OUT + col];
        c = __builtin_amdgcn_wmma_f32_16x16x4_f32(false, a, false, b,
                                                  (short)0, c, false, false);
    }

    const float bv = bias[col];
    #pragma unroll
    for (int v = 0; v < 8; ++v) {
        const int r = rowt * 16 + v + half * 8;
        float val = c[v] + bv;
        if (RELU) val = fmaxf(val, 0.0f);
        out[(size_t)r * OUT + col] = val;
    }
}

// ---------------------------------------------------------------------------
// Launch: layer1 (emb via x_indices) -> h, layer2 (h) -> d_out
// Workspace: agg[N*128] | h[N*128] | cnt[N]   (~103 MB)
// ---------------------------------------------------------------------------
extern "C" void kernel_launch(void* const* d_in, const int* in_sizes, int n_in,
                              void* d_out, int out_size, void* d_ws, size_t ws_size,
                              hipStream_t stream) {
    (void)in_sizes; (void)n_in; (void)out_size; (void)ws_size;

    const int*   xidx = (const int*)d_in[0];
    const int*   ei   = (const int*)d_in[1];
    const float* emb  = (const float*)d_in[2];
    const float* Wl1  = (const float*)d_in[3];
    const float* bl1  = (const float*)d_in[4];
    const float* Wr1  = (const float*)d_in[5];
    const float* Wl2  = (const float*)d_in[6];
    const float* bl2  = (const float*)d_in[7];
    const float* Wr2  = (const float*)d_in[8];

    float* agg = (float*)d_ws;
    float* h   = agg + (size_t)N_NODES * HIDDEN;
    float* cnt = h   + (size_t)N_NODES * HIDDEN;

    const int aggN4 = N_NODES * EMB_DIM / 4;   // 3.2M float4
    const int cntN4 = N_NODES / 4;             // 25k float4
    const int scatT = N_EDGES * (EMB_DIM / 4); // 25.6M threads

    dim3 wblk(32, 4);

    // --- Layer 1: agg/cnt from emb[x_indices] ---
    zero4_kernel<<<2048, 256, 0, stream>>>((float4*)agg, aggN4);
    zero4_kernel<<<128, 256, 0, stream>>>((float4*)cnt, cntN4);
    degree_kernel<<<(N_EDGES + 255) / 256, 256, 0, stream>>>(ei, cnt);
    scatter_kernel<EMB_DIM><<<(scatT + 255) / 256, 256, 0, stream>>>(ei, emb, xidx, agg);
    sage_wmma_kernel<EMB_DIM, HIDDEN, true>
        <<<dim3(N_NODES / 16, HIDDEN / 16 / 4), wblk, 0, stream>>>(
            agg, cnt, emb, xidx, Wl1, bl1, Wr1, h);

    // --- Layer 2: agg from h (cnt unchanged) ---
    zero4_kernel<<<2048, 256, 0, stream>>>((float4*)agg, aggN4);
    scatter_kernel<HIDDEN><<<(scatT + 255) / 256, 256, 0, stream>>>(ei, h, nullptr, agg);
    sage_wmma_kernel<HIDDEN, OUT_C, false>
        <<<dim3(N_NODES / 16, OUT_C / 16 / 4), wblk, 0, stream>>>(
            agg, cnt, h, nullptr, Wl2, bl2, Wr2, (float*)d_out);
}